// SelfAttention_14723147890812
// MI455X (gfx1250) — compile-verified
//
#include <hip/hip_runtime.h>
#include <hip/hip_bf16.h>
#include <math.h>

// ---------------------------------------------------------------------------
// Causal self-attention, T=8192, E=1024, D=11 (padded to 16), all f32.
// Flash-attention streaming with V_WMMA_F32_16X16X4_F32 tiles.
// ---------------------------------------------------------------------------

static constexpr int TSEQ = 8192;
static constexpr int EDIM = 1024;
static constexpr int DDIM = 11;
static constexpr int DP   = 16;   // padded head dim

typedef float v2f __attribute__((ext_vector_type(2)));
typedef float v8f __attribute__((ext_vector_type(8)));

// D = A(16x4) * B(4x16) + C(16x16), all f32, one wave.
__device__ __forceinline__ v8f wmma_k4(v2f a, v2f b, v8f c) {
#if defined(__gfx1250__)
  return __builtin_amdgcn_wmma_f32_16x16x4_f32(
      /*neg_a=*/false, a, /*neg_b=*/false, b,
      /*c_mod=*/(short)0, c, /*reuse_a=*/false, /*reuse_b=*/false);
#else
  // Host / non-gfx1250 pass: parse-only dummy (never executed).
  v8f d = c;
  d[0] += a.x * b.x + a.y * b.y;
  return d;
#endif
}

// ---------------------------------------------------------------------------
// Kernel 1: q/k/v = x_emb @ W{q,k,v}.T + b  -> padded [T, DP] workspace.
// One wave per 16-row tile. K-loop over E in steps of 4 (f32 WMMA).
// A layout (16x4 f32): lane<16 holds K={0,1}, lane>=16 holds K={2,3}, M=lane%16.
// B layout (4x16 f32): mirrors A with N=lane%16.
// Weight loads are branch-free: clamped row + cndmask-to-zero (EXEC untouched).
// ---------------------------------------------------------------------------
__global__ void qkv_proj_kernel(const float* __restrict__ xe,
                                const float* __restrict__ Wk, const float* __restrict__ bk,
                                const float* __restrict__ Wq, const float* __restrict__ bq,
                                const float* __restrict__ Wv, const float* __restrict__ bv,
                                float* __restrict__ qp, float* __restrict__ kp,
                                float* __restrict__ vp) {
  const int wave = threadIdx.x >> 5;
  const int lane = threadIdx.x & 31;
  const int half = lane >> 4;
  const int l16  = lane & 15;
  const int tile = blockIdx.x * (blockDim.x >> 5) + wave;
  if (tile >= TSEQ / 16) return;

  const float* xrow = xe + (size_t)(tile * 16 + l16) * EDIM;
  const bool wvalid = (l16 < DDIM);
  const int  wrow   = wvalid ? l16 : 0;           // clamped: address always valid
  const float* wkrow = Wk + (size_t)wrow * EDIM;
  const float* wqrow = Wq + (size_t)wrow * EDIM;
  const float* wvrow = Wv + (size_t)wrow * EDIM;
  const v2f zero2 = {0.0f, 0.0f};

  v8f cq = {}, ck = {}, cv = {};
  for (int e = 0; e < EDIM; e += 4) {
    const int ka = e + 2 * half;                  // this lane's K pair base
    v2f a = *(const v2f*)(xrow + ka);
    v2f tk = *(const v2f*)(wkrow + ka);           // unconditional b64 loads
    v2f tq = *(const v2f*)(wqrow + ka);
    v2f tv = *(const v2f*)(wvrow + ka);
    v2f bkv = wvalid ? tk : zero2;                // v_cndmask, no EXEC games
    v2f bqv = wvalid ? tq : zero2;
    v2f bvv = wvalid ? tv : zero2;
    ck = wmma_k4(a, bkv, ck);
    cq = wmma_k4(a, bqv, cq);
    cv = wmma_k4(a, bvv, cv);
  }

  const float biask = wvalid ? bk[wrow] : 0.0f;
  const float biasq = wvalid ? bq[wrow] : 0.0f;
  const float biasv = wvalid ? bv[wrow] : 0.0f;

#pragma unroll
  for (int r = 0; r < 8; ++r) {
    const int m = r + 8 * half;                   // C layout: row = r + 8*half
    const size_t off = (size_t)(tile * 16 + m) * DP + l16;
    kp[off] = ck[r] + biask;
    qp[off] = cq[r] + biasq;
    vp[off] = cv[r] + biasv;
  }
}

// ---------------------------------------------------------------------------
// Kernel 2: streaming causal attention. One wave = one 16-row query tile.
// For each key tile j<=i:  S = Q Kj^T (4 WMMA), online softmax in C-layout
// (shfl_xor row reductions), P staged via LDS to A-layout, O += P Vj (4 WMMA).
// K/V padded tiles are tiny (512 KB each) -> L2 resident on the 192MB L2;
// next tile is prefetched (global_prefetch_b8) while this one computes.
// ---------------------------------------------------------------------------
__global__ void flash_attn_kernel(const float* __restrict__ qp,
                                  const float* __restrict__ kp,
                                  const float* __restrict__ vp,
                                  float* __restrict__ rp) {
  __shared__ float plds[4 * 256];                 // 4 waves/block, 16x16 f32 each
  const int wave = threadIdx.x >> 5;
  const int lane = threadIdx.x & 31;
  const int half = lane >> 4;
  const int l16  = lane & 15;
  float* myP = plds + wave * 256;

  const int it = blockIdx.x * (blockDim.x >> 5) + wave;
  if (it >= TSEQ / 16) return;

  // Preload Q tile in A layout: 4 K-chunks of 4.
  const float* qrow = qp + (size_t)(it * 16 + l16) * DP;
  v2f aq[4];
#pragma unroll
  for (int kb = 0; kb < 4; ++kb)
    aq[kb] = *(const v2f*)(qrow + kb * 4 + 2 * half);

  v8f o = {};
  float mrun[8], lrun[8];
#pragma unroll
  for (int r = 0; r < 8; ++r) { mrun[r] = -INFINITY; lrun[r] = 0.0f; }

  for (int jt = 0; jt <= it; ++jt) {
    // ---- prefetch next key/value tile into cache (uniform branch)
    if (jt < it) {
      __builtin_prefetch(kp + (size_t)((jt + 1) * 16 + l16) * DP, 0, 3);
      __builtin_prefetch(vp + (size_t)((jt + 1) * 16 + l16) * DP, 0, 3);
    }

    // ---- S = Q @ Kj^T : B[k][n] = kp[jt*16+n][k] (contiguous b64 per lane)
    const float* krow = kp + (size_t)(jt * 16 + l16) * DP;
    v8f s = {};
#pragma unroll
    for (int kb = 0; kb < 4; ++kb) {
      v2f b = *(const v2f*)(krow + kb * 4 + 2 * half);
      s = wmma_k4(aq[kb], b, s);
    }

    // ---- causal mask on the diagonal tile (select, no divergence)
    if (jt == it) {
#pragma unroll
      for (int r = 0; r < 8; ++r) {
        const int m = r + 8 * half;
        s[r] = (l16 > m) ? -INFINITY : s[r];
      }
    }

    // ---- online softmax per row (row = 16-lane group in C layout)
#pragma unroll
    for (int r = 0; r < 8; ++r) {
      float x = s[r];
#pragma unroll
      for (int off = 8; off >= 1; off >>= 1)
        x = fmaxf(x, __shfl_xor(x, off));         // masks <16: stays in row group
      const float mnew  = fmaxf(mrun[r], x);
      const float alpha = __expf(mrun[r] - mnew);
      const float pe    = __expf(s[r] - mnew);    // masked entries -> exp(-inf)=0
      float psum = pe;
#pragma unroll
      for (int off = 8; off >= 1; off >>= 1)
        psum += __shfl_xor(psum, off);
      lrun[r] = lrun[r] * alpha + psum;
      mrun[r] = mnew;
      o[r] *= alpha;
      myP[(r + 8 * half) * 16 + l16] = pe;        // P in row-major LDS
    }

    // same-wave DS ordering + explicit wait before cross-lane re-read
    asm volatile("s_wait_dscnt 0" ::: "memory");

    // ---- O += P @ Vj : A from LDS (transposed read), B[k][d] = vp[jt*16+k][d]
    const float* vbase = vp + (size_t)(jt * 16) * DP;
#pragma unroll
    for (int kb = 0; kb < 4; ++kb) {
      const int k0 = kb * 4 + 2 * half;
      v2f ap = *(const v2f*)(myP + l16 * 16 + k0);    // ds_load_b64
      v2f bv2;
      bv2.x = vbase[(size_t)(k0 + 0) * DP + l16];
      bv2.y = vbase[(size_t)(k0 + 1) * DP + l16];
      o = wmma_k4(ap, bv2, o);
    }
  }

  // ---- normalize and store padded result tile
#pragma unroll
  for (int r = 0; r < 8; ++r) {
    const int m = r + 8 * half;
    rp[(size_t)(it * 16 + m) * DP + l16] = o[r] / lrun[r];
  }
}

// ---------------------------------------------------------------------------
// Kernel 3: out = res @ Wf.T + bf.  One wave per 16x16 output tile.
// K runs over D=11 (chunks kb=0..2 cover k<12; chunk 3 is all-zero: skipped).
// Wf loads are branch-free: clamped k + cndmask-to-zero.
// ---------------------------------------------------------------------------
__global__ void out_proj_kernel(const float* __restrict__ rp,
                                const float* __restrict__ Wf,
                                const float* __restrict__ bf,
                                float* __restrict__ out) {
  const int wave = threadIdx.x >> 5;
  const int lane = threadIdx.x & 31;
  const int half = lane >> 4;
  const int l16  = lane & 15;

  const int gw  = blockIdx.x * (blockDim.x >> 5) + wave;
  const int nct = EDIM / 16;                      // 64 column tiles
  const int it  = gw / nct;
  const int jt  = gw % nct;
  if (it >= TSEQ / 16) return;

  const float* rrow = rp + (size_t)(it * 16 + l16) * DP;
  const int n = jt * 16 + l16;                    // global output column
  const float* wfrow = Wf + (size_t)n * DDIM;

  v8f c = {};
#pragma unroll
  for (int kb = 0; kb < 3; ++kb) {
    const int k0 = kb * 4 + 2 * half;
    v2f a = *(const v2f*)(rrow + k0);
    // clamped (always-valid) loads + selects; only k0=10 hits the k=11 edge
    const int k0c = (k0     < DDIM) ? k0     : 0;
    const int k1c = (k0 + 1 < DDIM) ? k0 + 1 : 0;
    const float w0 = wfrow[k0c];
    const float w1 = wfrow[k1c];
    v2f b;
    b.x = (k0     < DDIM) ? w0 : 0.0f;
    b.y = (k0 + 1 < DDIM) ? w1 : 0.0f;
    c = wmma_k4(a, b, c);
  }

  const float bias = bf[n];
#pragma unroll
  for (int r = 0; r < 8; ++r) {
    const int m = r + 8 * half;
    out[(size_t)(it * 16 + m) * EDIM + n] = c[r] + bias;
  }
}

// ---------------------------------------------------------------------------
extern "C" void kernel_launch(void* const* d_in, const int* in_sizes, int n_in,
                              void* d_out, int out_size, void* d_ws, size_t ws_size,
                              hipStream_t stream) {
  // inputs: 0:x(int64,unused) 1:x_embeddings 2:Wk 3:bk 4:Wq 5:bq 6:Wv 7:bv 8:Wf 9:bf
  const float* xe = (const float*)d_in[1];
  const float* Wk = (const float*)d_in[2];
  const float* bk = (const float*)d_in[3];
  const float* Wq = (const float*)d_in[4];
  const float* bq = (const float*)d_in[5];
  const float* Wv = (const float*)d_in[6];
  const float* bv = (const float*)d_in[7];
  const float* Wf = (const float*)d_in[8];
  const float* bf = (const float*)d_in[9];
  float* out = (float*)d_out;

  float* ws = (float*)d_ws;
  float* qp = ws + (size_t)0 * TSEQ * DP;         // [T, 16] padded
  float* kp = ws + (size_t)1 * TSEQ * DP;
  float* vp = ws + (size_t)2 * TSEQ * DP;
  float* rp = ws + (size_t)3 * TSEQ * DP;

  const int waves_per_block = 4;
  const int threads = waves_per_block * 32;

  const int qkv_tiles = TSEQ / 16;                               // 512
  qkv_proj_kernel<<<qkv_tiles / waves_per_block, threads, 0, stream>>>(
      xe, Wk, bk, Wq, bq, Wv, bv, qp, kp, vp);

  flash_attn_kernel<<<qkv_tiles / waves_per_block, threads, 0, stream>>>(
      qp, kp, vp, rp);

  const int out_tiles = (TSEQ / 16) * (EDIM / 16);               // 32768
  out_proj_kernel<<<out_tiles / waves_per_block, threads, 0, stream>>>(
      rp, Wf, bf, out);
}